// QuantumReservoirFeedbackTimeSeries_4999341932813
// MI455X (gfx1250) — compile-verified
//
#include <hip/hip_runtime.h>
#include <cstdint>

// QuantumReservoirFeedbackTimeSeries — closed-form parallel rewrite.
// out[b,t] = f(x[b,t-5..t]); pure streaming stencil, store-BW bound.

#define MEM_DEPTH 5
#define TPB       256          // 8 wave32 per workgroup
#define VPT       8            // timesteps per thread
#define TILE_T    (TPB * VPT)  // 2048 timesteps per workgroup
#define PI_F      3.14159265358979323846f

// ---- CDNA5 async global->LDS copies (CDNA5 ISA) ----------
__device__ __forceinline__ void async_load_lds_b128(uint32_t lds_byte_off,
                                                    const void* gaddr) {
#if defined(__gfx1250__)
  asm volatile("global_load_async_to_lds_b128 %0, %1, off"
               :: "v"(lds_byte_off), "v"((uint64_t)(uintptr_t)gaddr)
               : "memory");
#endif
}

__device__ __forceinline__ void async_load_lds_b32(uint32_t lds_byte_off,
                                                   const void* gaddr) {
#if defined(__gfx1250__)
  asm volatile("global_load_async_to_lds_b32 %0, %1, off"
               :: "v"(lds_byte_off), "v"((uint64_t)(uintptr_t)gaddr)
               : "memory");
#endif
}

__device__ __forceinline__ void wait_asynccnt0() {
#if defined(__gfx1250__)
  asm volatile("s_wait_asynccnt 0x0" ::: "memory");
#endif
}

__global__ __launch_bounds__(TPB)
void qres_feedback_kernel(const float* __restrict__ x,
                          float* __restrict__ out,
                          int S) {
  // xs[3..7]        : 5-float halo  = x[tile_start-5 .. tile_start-1]
  // xs[8..8+2047]   : main tile     = x[tile_start   .. tile_start+2047]
  __shared__ __align__(16) float xs[TILE_T + 8];

  const int b          = blockIdx.y;
  const int tile_start = blockIdx.x * TILE_T;
  const int tid        = threadIdx.x;
  const float* __restrict__ xrow = x + (size_t)b * (size_t)S;

  const uint32_t lds_base = (uint32_t)(uintptr_t)xs;  // low 32 bits = LDS offset

  // Main tile: 2048 floats as 512 async b128 loads, 2 per thread, coalesced.
#pragma unroll
  for (int r = 0; r < 2; ++r) {
    const int v4 = r * TPB + tid;                       // vec4 index 0..511
    async_load_lds_b128(lds_base + (uint32_t)(8 + v4 * 4) * 4u,
                        xrow + tile_start + v4 * 4);
  }
  // Halo: 5 async b32 loads (skipped for the first tile; c<0 forced to 0 below).
  if (tile_start > 0 && tid < MEM_DEPTH) {
    async_load_lds_b32(lds_base + (uint32_t)(3 + tid) * 4u,
                       xrow + tile_start - MEM_DEPTH + tid);
  }
  wait_asynccnt0();
  __syncthreads();

  // Per-thread: 13 c-values cover 8 outputs + 5-tap left window.
  const int base_j = tid * VPT;
  float c[VPT + MEM_DEPTH];
#pragma unroll
  for (int j = 0; j < VPT + MEM_DEPTH; ++j) {
    const int   gt = tile_start + base_j + j - MEM_DEPTH;   // global timestep
    const float cv = 0.5f + 0.5f * __cosf(PI_F * xs[3 + base_j + j]);
    c[j] = (gt >= 0) ? cv : 0.0f;     // select (not mul) so garbage LDS is safe
  }

  // Sliding 5-tap window sum; buffer divisor is always MEM_DEPTH.
  float s = c[0] + c[1] + c[2] + c[3] + c[4];
  float op[3 * VPT];
#pragma unroll
  for (int i = 0; i < VPT; ++i) {
    const float p0   = c[i + MEM_DEPTH];            // cos^2(phi/2)
    const float sphi = 1.0f - p0;                   // sin^2(phi/2)
    const float R    = 0.2f * s;                    // mean of last 5 p0's
    const float e    = __expf(-2.0f * R);           // v_exp_f32
    const float th   = (1.0f - e) * __builtin_amdgcn_rcpf(1.0f + e); // tanh(R)
    const float sth  = 0.5f - 0.5f * __cosf(PI_F * th);  // sin^2(theta/2)
    const float p1   = sphi * sth;
    op[3 * i + 0] = p0;
    op[3 * i + 1] = p1;
    op[3 * i + 2] = sphi - p1;
    s += p0 - c[i];                                 // slide window
  }

  // 24 contiguous floats per thread -> six aligned float4 stores.
  // (b*S + t0) is a multiple of 8  =>  byte offset multiple of 96, 16B aligned.
  float4* __restrict__ dst =
      (float4*)(out + ((size_t)b * (size_t)S + (size_t)(tile_start + base_j)) * 3);
#pragma unroll
  for (int k = 0; k < (3 * VPT) / 4; ++k) {
    dst[k] = make_float4(op[4 * k + 0], op[4 * k + 1],
                         op[4 * k + 2], op[4 * k + 3]);
  }
}

extern "C" void kernel_launch(void* const* d_in, const int* in_sizes, int n_in,
                              void* d_out, int out_size, void* d_ws, size_t ws_size,
                              hipStream_t stream) {
  (void)n_in; (void)out_size; (void)d_ws; (void)ws_size;
  const float* x  = (const float*)d_in[0];
  float*      out = (float*)d_out;

  const int S = 8192;                 // reference: x is [512, 8192, 1] f32
  const int B = in_sizes[0] / S;      // = 512

  dim3 grid(S / TILE_T, B);           // (4, 512) workgroups
  qres_feedback_kernel<<<grid, TPB, 0, stream>>>(x, out, S);
}